// SelfMaskedAttention_38903813767697
// MI455X (gfx1250) — compile-verified
//
#include <hip/hip_runtime.h>
#include <math.h>
#include <stdint.h>

// CDNA5 fp32 WMMA: D(16x16) = A(16x4) * B(4x16) + C. 2 VGPRs A/B, 8 VGPRs C/D.
typedef __attribute__((ext_vector_type(2))) float v2f;
typedef __attribute__((ext_vector_type(8))) float v8f;

__device__ __forceinline__ v8f wmma4(v2f a, v2f b, v8f c) {
  // (neg_a, A, neg_b, B, c_mod, C, reuse_a, reuse_b)
  return __builtin_amdgcn_wmma_f32_16x16x4_f32(false, a, false, b, (short)0, c,
                                               false, false);
}

// gfx1250 async global->LDS copy (VGLOBAL GLOBAL_LOAD_ASYNC_TO_LDS_B128,
// tracked by ASYNCcnt). LDS operand is the 32-bit LDS byte offset (generic
// pointers to LDS carry the offset in their low 32 bits per the aperture map).
__device__ __forceinline__ void async_b128(void* lds, const void* gaddr) {
  uint32_t loff = (uint32_t)(uintptr_t)lds;
  asm volatile("global_load_async_to_lds_b128 %0, %1, off"
               :: "v"(loff), "v"(gaddr)
               : "memory");
}
__device__ __forceinline__ void wait_async0() {
  asm volatile("s_wait_asynccnt 0x0" ::: "memory");
}

// ---------------------------------------------------------------------------
// 128x64-tile fp32 WMMA GEMM, double-buffered async-LDS staging.
// C = A(MxK) @ B(KxN) + bias.  8 waves; each wave: 64(M) x 16(N) = 4 subtiles
// sharing one B fragment (2.5 LDS dwords per WMMA).
// mode 0: plain row-major store to C.
// mode 1: QKV scatter epilogue: cols [0,768) -> q_ws (B,H,T,DH)
//                               cols [768,1536) -> kdst (B,H,DH,T)  (k_out)
//                               cols [1536,2304) -> vdst (B,H,T,DH) (v out)
// ---------------------------------------------------------------------------
__global__ __launch_bounds__(256)
void gemm128_kernel(const float* __restrict__ A, const float* __restrict__ Bm,
                    const float* __restrict__ bias, float* __restrict__ C,
                    float* __restrict__ qdst, float* __restrict__ kdst,
                    float* __restrict__ vdst, int M, int N, int K, int mode) {
  // pitch 36 floats: 16B-aligned rows for async B128; stride 36 mod 64 banks
  // are distinct for 16 lanes, and the +2 half-lane column is disjoint mod 4.
  __shared__ float As[2][128][36];
  __shared__ float Bs[2][32][64];
  const int tid  = threadIdx.x;
  const int w    = tid >> 5;
  const int lane = tid & 31;
  const int hl   = lane >> 4;      // lane half selects K pair {0,1} vs {2,3}
  const int lr   = lane & 15;
  const int n0   = (w & 3) * 16;
  const int mh   = (w >> 2) * 64;  // 0 or 64
  const int gm0  = blockIdx.y * 128;
  const int gn0  = blockIdx.x * 64;
  v8f acc[4] = {{}, {}, {}, {}};

  auto prefetch = [&](int buf, int k0) {
    for (int i = tid; i < 1024; i += 256) {            // A tile 128x32
      int m = i >> 3, c4 = (i & 7) * 4;
      async_b128(&As[buf][m][c4], A + (size_t)(gm0 + m) * K + k0 + c4);
    }
    for (int i = tid; i < 512; i += 256) {             // B tile 32x64
      int kk = i >> 4, c4 = (i & 15) * 4;
      async_b128(&Bs[buf][kk][c4], Bm + (size_t)(k0 + kk) * N + gn0 + c4);
    }
  };

  prefetch(0, 0);
  int cur = 0;
  for (int k0 = 0; k0 < K; k0 += 32) {
    wait_async0();            // my async loads for buf[cur] are in LDS
    __syncthreads();          // everyone's are; everyone done reading buf[cur^1]
    if (k0 + 32 < K) prefetch(cur ^ 1, k0 + 32);
    for (int kk = 0; kk < 32; kk += 4) {
      v2f b;
      b.x = Bs[cur][kk + 2 * hl][n0 + lr];
      b.y = Bs[cur][kk + 2 * hl + 1][n0 + lr];
#pragma unroll
      for (int p = 0; p < 4; ++p) {
        v2f a;
        a.x = As[cur][mh + p * 16 + lr][kk + 2 * hl];
        a.y = As[cur][mh + p * 16 + lr][kk + 2 * hl + 1];
        acc[p] = wmma4(a, b, acc[p]);
      }
    }
    cur ^= 1;
  }

  for (int part = 0; part < 4; ++part) {
    int m0 = mh + part * 16;
    for (int r = 0; r < 8; ++r) {
      int row = gm0 + m0 + r + 8 * hl;   // C/D layout: vgpr r -> M=r / r+8
      int col = gn0 + n0 + lr;
      float val = acc[part][r] + bias[col];
      if (mode == 0) {
        C[(size_t)row * N + col] = val;
      } else {
        int b = row >> 9, t = row & 511; // row = b*512 + t
        if (col < 768) {
          int d = col;
          qdst[(((size_t)b * 12 + (d >> 6)) * 512 + t) * 64 + (d & 63)] = val;
        } else if (col < 1536) {
          int d = col - 768;             // k_out layout (B,H,DH,T)
          kdst[(((size_t)b * 12 + (d >> 6)) * 64 + (d & 63)) * 512 + t] = val;
        } else {
          int d = col - 1536;
          vdst[(((size_t)b * 12 + (d >> 6)) * 512 + t) * 64 + (d & 63)] = val;
        }
      }
    }
  }
}

// biasTab[h][p] = sum_d (content_bias[h,d] + position_bias[h,d]) * proj[p,h,d]
__global__ void bias_tab_kernel(const float* __restrict__ proj,
                                const float* __restrict__ cb,
                                const float* __restrict__ pb,
                                float* __restrict__ tab) {
  int i = blockIdx.x * blockDim.x + threadIdx.x;
  if (i >= 12 * 512) return;
  int h = i >> 9, p = i & 511;
  const float* pr = proj + (size_t)p * 768 + h * 64;
  float s = 0.f;
  for (int d = 0; d < 64; ++d) s += (cb[h * 64 + d] + pb[h * 64 + d]) * pr[d];
  tab[i] = s;
}

// ---------------------------------------------------------------------------
// Fused causal attention per (b,h, 32-query block). 128 threads = 4 waves.
// scores = q@k^T + gather(q@proj^T, t-s) + biasSum[t-s]; e = exp(s/9)*mask;
// ctx = (sum_s e * v) / (sum_s e + 2).   No running-max needed (plain exp).
// All tile staging via async global->LDS.
// ---------------------------------------------------------------------------
__global__ __launch_bounds__(128)
void attn_kernel(const float* __restrict__ q, const float* __restrict__ k,
                 const float* __restrict__ v, const float* __restrict__ proj,
                 const float* __restrict__ btab, float* __restrict__ ctx) {
  __shared__ float Qs[32][68];   // pitch 68: 16B rows + conflict-free A reads
  __shared__ float Ks[64][32];   // k tile as (dh x s) == B operand row-major
  __shared__ float Vs[32][64];   // v tile (s x dh)  == B operand row-major
  __shared__ float Ps[64][68];   // proj window (p x dh)
  __shared__ float Rs[32][65];   // rel tile R[t][p-pbase]
  __shared__ float Es[32][33];   // exp tile (A-operand reads, 33 coprime 64)
  __shared__ float Bsum[64];
  __shared__ float rowsum[32];

  const int tid  = threadIdx.x;
  const int bh   = blockIdx.y;
  const int b    = bh / 12;
  const int h    = bh % 12;
  const int t0   = blockIdx.x * 32;
  const int w    = tid >> 5;
  const int lane = tid & 31;
  const int hl   = lane >> 4;
  const int lr   = lane & 15;
  const int m0   = (w & 1) * 16;     // this wave's 16 query rows
  const int n0s  = (w >> 1) * 16;    // score subtile N offset
  const int n0c  = (w >> 1) * 32;    // ctx N offset (two subtiles)
  const float inv9 = 1.0f / 9.0f;    // 1/(sqrt(64)+EPS)

  const float* qb = q + ((size_t)bh * 512 + t0) * 64;
  for (int i = tid; i < 512; i += 128) {               // Q tile 32x64 (async)
    int r = i >> 4, c4 = (i & 15) * 4;
    async_b128(&Qs[r][c4], qb + r * 64 + c4);
  }
  if (tid < 32) rowsum[tid] = 0.f;
  v8f cacc0 = {};
  v8f cacc1 = {};

  for (int s0 = 0; s0 <= t0; s0 += 32) {
    __syncthreads();                                   // protect tile reuse
    const float* kb = k + (size_t)bh * 64 * 512 + s0;  // (B,H,DH,T) rows = dh
    for (int i = tid; i < 512; i += 128) {
      int dh = i >> 3, c4 = (i & 7) * 4;
      async_b128(&Ks[dh][c4], kb + (size_t)dh * 512 + c4);
    }
    const float* vb = v + ((size_t)bh * 512 + s0) * 64;
    for (int i = tid; i < 512; i += 128) {
      int s = i >> 4, c4 = (i & 15) * 4;
      async_b128(&Vs[s][c4], vb + s * 64 + c4);
    }
    int pbase = t0 - s0 - 31; if (pbase < 0) pbase = 0;  // p=t-s window start
    for (int i = tid; i < 1024; i += 128) {
      int j = i >> 4, c4 = (i & 15) * 4;
      int p = pbase + j;
      if (p < 512) {
        async_b128(&Ps[j][c4], proj + (size_t)p * 768 + h * 64 + c4);
      } else {
        Ps[j][c4] = 0.f; Ps[j][c4 + 1] = 0.f; Ps[j][c4 + 2] = 0.f; Ps[j][c4 + 3] = 0.f;
      }
    }
    if (tid < 64) {
      int p = pbase + tid;
      Bsum[tid] = (p < 512) ? btab[h * 512 + p] : 0.f;
    }
    wait_async0();
    __syncthreads();

    // --- scores subtile: q @ k^T over dh (16 WMMA steps of K=4) ---
    v8f sacc = {};
    for (int kk = 0; kk < 64; kk += 4) {
      v2f a;  a.x  = Qs[m0 + lr][kk + 2 * hl];  a.y  = Qs[m0 + lr][kk + 2 * hl + 1];
      v2f bb; bb.x = Ks[kk + 2 * hl][n0s + lr]; bb.y = Ks[kk + 2 * hl + 1][n0s + lr];
      sacc = wmma4(a, bb, sacc);
    }
    // --- relative term tile: R[t][j] = q[t] . proj[pbase+j] (2 subtiles/wave)
    const int j0 = (w >> 1) * 32;
    for (int part = 0; part < 2; ++part) {
      int jj = j0 + part * 16;
      v8f racc = {};
      for (int kk = 0; kk < 64; kk += 4) {
        v2f a;  a.x  = Qs[m0 + lr][kk + 2 * hl];  a.y  = Qs[m0 + lr][kk + 2 * hl + 1];
        v2f bb; bb.x = Ps[jj + lr][kk + 2 * hl]; bb.y = Ps[jj + lr][kk + 2 * hl + 1];
        racc = wmma4(a, bb, racc);
      }
      for (int r = 0; r < 8; ++r) Rs[m0 + r + 8 * hl][jj + lr] = racc[r];
    }
    __syncthreads();

    // --- epilogue: gather rel + bias, scale, masked exp -> Es ---
    for (int r = 0; r < 8; ++r) {
      int m = m0 + r + 8 * hl;
      int n = n0s + lr;
      int t = t0 + m, s = s0 + n;
      float e = 0.f;
      if (s <= t) {
        int pi = (t - s) - pbase;                      // in [0,62]
        float val = (sacc[r] + Rs[m][pi] + Bsum[pi]) * inv9;
        e = __expf(val);
      }
      Es[m][n] = e;
    }
    __syncthreads();

    if (tid < 32) {                                    // running denominator
      float rs = 0.f;
      for (int n = 0; n < 32; ++n) rs += Es[tid][n];
      rowsum[tid] += rs;
    }
    // --- ctx accumulation: Es(32x32) @ Vs(32x64), persistent accumulators ---
    for (int kk = 0; kk < 32; kk += 4) {
      v2f a;  a.x  = Es[m0 + lr][kk + 2 * hl];       a.y  = Es[m0 + lr][kk + 2 * hl + 1];
      v2f b0; b0.x = Vs[kk + 2 * hl][n0c + lr];      b0.y = Vs[kk + 2 * hl + 1][n0c + lr];
      v2f b1; b1.x = Vs[kk + 2 * hl][n0c + 16 + lr]; b1.y = Vs[kk + 2 * hl + 1][n0c + 16 + lr];
      cacc0 = wmma4(a, b0, cacc0);
      cacc1 = wmma4(a, b1, cacc1);
    }
  }
  __syncthreads();

  // normalize and scatter into (B,T, dh*H + h) interleaved ctx layout
  for (int r = 0; r < 8; ++r) {
    int m = m0 + r + 8 * hl;
    int t = t0 + m;
    float denom = rowsum[m] + 2.0f;                    // + 2*EPS
    size_t base = ((size_t)b * 512 + t) * 768 + h;
    ctx[base + (size_t)(n0c + lr) * 12]      = cacc0[r] / denom;
    ctx[base + (size_t)(n0c + 16 + lr) * 12] = cacc1[r] / denom;
  }
}

// ---------------------------------------------------------------------------
extern "C" void kernel_launch(void* const* d_in, const int* in_sizes, int n_in,
                              void* d_out, int out_size, void* d_ws, size_t ws_size,
                              hipStream_t stream) {
  (void)in_sizes; (void)n_in; (void)out_size; (void)ws_size;
  const float* x    = (const float*)d_in[0];
  const float* rpe  = (const float*)d_in[1];
  const float* rcb  = (const float*)d_in[2];
  const float* rpb  = (const float*)d_in[3];
  const float* Wqkv = (const float*)d_in[4];
  const float* bqkv = (const float*)d_in[5];
  const float* Wh   = (const float*)d_in[6];
  const float* bhb  = (const float*)d_in[7];
  const float* Wp   = (const float*)d_in[8];
  const float* bp   = (const float*)d_in[9];

  float* out  = (float*)d_out;                   // (B,T,D)
  float* kout = out + (size_t)8 * 512 * 768;     // (B,H,DH,T)
  float* vout = kout + (size_t)8 * 512 * 768;    // (B,H,T,DH)

  float* ws   = (float*)d_ws;                    // ~27 MB fp32 scratch
  float* q_ws = ws;                              // 3,145,728
  float* proj = ws + 3145728;                    // 393,216  (P=512, D=768)
  float* btab = proj + 393216;                   // 6,144    (H x T)
  float* ctx  = btab + 6144;                     // 3,145,728

  // 1) QKV projection + head split/scatter (k,v land directly in d_out)
  gemm128_kernel<<<dim3(36, 32), 256, 0, stream>>>(x, Wqkv, bqkv, nullptr,
                                                   q_ws, kout, vout,
                                                   4096, 2304, 768, 1);
  // 2) relative position projection
  gemm128_kernel<<<dim3(12, 4), 256, 0, stream>>>(rpe, Wp, bp, proj,
                                                  nullptr, nullptr, nullptr,
                                                  512, 768, 768, 0);
  // 3) combined content/position bias table
  bias_tab_kernel<<<24, 256, 0, stream>>>(proj, rcb, rpb, btab);
  // 4) fused relative-position causal attention
  attn_kernel<<<dim3(16, 96), 128, 0, stream>>>(q_ws, kout, vout, proj, btab, ctx);
  // 5) output projection
  gemm128_kernel<<<dim3(12, 32), 256, 0, stream>>>(ctx, Wh, bhb, out,
                                                   nullptr, nullptr, nullptr,
                                                   4096, 768, 768, 0);
}